// kernel_generated_0_21887153341320
// MI455X (gfx1250) — compile-verified
//
#include <hip/hip_runtime.h>

typedef __attribute__((ext_vector_type(2))) float v2f;
typedef __attribute__((ext_vector_type(8))) float v8f;
typedef __attribute__((ext_vector_type(4))) int   v4i;

#define HW   56
#define HW2  3136            // 56*56
#define XROW 60              // padded LDS row stride in floats (240B, 16B-aligned)

// Async global->LDS (gfx1250 ASYNCcnt path) with sync fallback.
// Builtin signature (probe-confirmed via diagnostic): (AS1 v4i*, AS3 v4i*, imm offset, imm cpol)
__device__ __forceinline__ void g2l_b128(float* ldst, const float* gsrc) {
#if __has_builtin(__builtin_amdgcn_global_load_async_to_lds_b128)
    __builtin_amdgcn_global_load_async_to_lds_b128(
        (__attribute__((address_space(1))) v4i*)gsrc,
        (__attribute__((address_space(3))) v4i*)ldst, 0, 0);
#else
    *(float4*)ldst = *(const float4*)gsrc;
#endif
}

__device__ __forceinline__ void wait_async_lds() {
#if __has_builtin(__builtin_amdgcn_s_wait_asynccnt)
    __builtin_amdgcn_s_wait_asynccnt(0);
#else
    asm volatile("s_wait_asynccnt 0" ::: "memory");
#endif
}

// One block per output row m. 128 threads = 4 waves (wave32).
// Stage 1: per wave, one 16-column tile; two K=192 GEMMs (o1=0/1) sharing A,
//          via V_WMMA_F32_16X16X4_F32 (full fp32), operands from LDS.
// Stage 2: one output channel per thread, 6 FMAs per pixel from LDS t4.
__global__ __launch_bounds__(128) void shift_conv_fused(
    const float* __restrict__ x,    // (128, 56, 56)
    const float* __restrict__ w1,   // (4, 64, 3)
    const float* __restrict__ w2,   // (32, 2, 3)
    float* __restrict__ out)        // (128, 56, 56)
{
    __shared__ __align__(16) float xs[2 * 192 * XROW]; // [o1][th=i1*3+j1][n] windowed x
    __shared__ __align__(16) float w1_lds[4 * 192];    // w1 as dense A (4 x 192)
    __shared__ float t4_lds[8][64];                    // [k1*2+o1][n]
    __shared__ float w2_lds[192];

    const int m    = blockIdx.x;
    const int tid  = threadIdx.x;
    const int wave = tid >> 5;
    const int lane = tid & 31;

    // ---- one-time fills (async b128 where possible) ----
    for (int q = tid; q < 192; q += 128) {             // w1: 768 floats = 192 x 4
        g2l_b128(&w1_lds[q * 4], &w1[q * 4]);
        w2_lds[q] = w2[q];                             // w2: plain cached load
    }

    // xs: 384 rows x 56 floats, roll(+1 over H) + H-tap zero-padding resolved here.
    //   row = o1*192 + th, th = i1*3 + j1, channel = 2*i1 + o1
    for (int q = tid; q < 384 * 14; q += 128) {
        const int row = q / 14;
        const int c16 = q - row * 14;                  // 16B chunk within row
        const int o1  = (row >= 192) ? 1 : 0;
        const int th  = row - 192 * o1;
        const int i1  = th / 3;
        const int j1  = th - 3 * i1;
        const int hm  = m + j1 - 1;                    // pre-roll H index
        float* dst = &xs[row * XROW + c16 * 4];
        if (hm >= 0 && hm < HW) {
            const int src_h = (hm - 1 + HW) % HW;      // roll by +1
            g2l_b128(dst, &x[(2 * i1 + o1) * HW2 + src_h * HW + c16 * 4]);
        } else {
            *(float4*)dst = make_float4(0.f, 0.f, 0.f, 0.f);
        }
    }
    wait_async_lds();
    __syncthreads();

    // ---- Stage 1: branchless WMMA loop, all operands in LDS ----
    const int r  = lane & 15;                          // A/D row id (rows 0..3 live)
    const int hi = lane >> 4;                          // K sub-block within 4-wide step
    const int nn = lane & 15;                          // B/D column within tile
    const int n0 = (wave < 3) ? wave * 16 : 40;        // tiles 0,16,32,40 (40 overlaps)

    const float* w1row = &w1_lds[(r & 3) * 192];
    const float* b0p   = &xs[n0 + nn];                 // o1 = 0 block
    const float* b1p   = &xs[192 * XROW + n0 + nn];    // o1 = 1 block

    v8f acc0 = {}, acc1 = {};
    #pragma unroll 4
    for (int s = 0; s < 48; ++s) {
        const int th0 = 4 * s + 2 * hi;                // K rows th0, th0+1 for this lane
        v2f a = *(const v2f*)(w1row + th0);            // dense A = w1 reshaped (4,192)
        if (r >= 4) a = (v2f){0.0f, 0.0f};             // M rows 4..15 are zero
        v2f b0, b1;
        b0[0] = b0p[(th0 + 0) * XROW];
        b0[1] = b0p[(th0 + 1) * XROW];
        b1[0] = b1p[(th0 + 0) * XROW];
        b1[1] = b1p[(th0 + 1) * XROW];
        acc0 = __builtin_amdgcn_wmma_f32_16x16x4_f32(false, a, false, b0,
                                                     (short)0, acc0, false, false);
        acc1 = __builtin_amdgcn_wmma_f32_16x16x4_f32(false, a, false, b1,
                                                     (short)0, acc1, false, false);
    }

    // D layout: lanes 0-15 hold rows 0..3 in VGPRs 0..3 (col = lane).
    if (lane < 16 && (wave < 3 || nn >= 8)) {          // tile 3 stores only cols 48..55
        const int n = n0 + nn;
        #pragma unroll
        for (int v = 0; v < 4; ++v) {
            t4_lds[v * 2 + 0][n] = acc0[v];            // t4[k1=v][o1=0]
            t4_lds[v * 2 + 1][n] = acc1[v];            // t4[k1=v][o1=1]
        }
    }
    __syncthreads();

    // ---- Stage 2: out[C=i2*4+o4][m][n] = sum_{j2<2,k3<3} t4[o4*2+j2][n+k3-1]*w2[i2][j2][k3]
    {
        const int c  = tid;                            // 128 threads -> 128 channels
        const int i2 = c >> 2;
        const int o4 = c & 3;
        float wv[6];
        #pragma unroll
        for (int u = 0; u < 6; ++u) wv[u] = w2_lds[i2 * 6 + u];
        const float* ta = &t4_lds[o4 * 2 + 0][0];
        const float* tb = &t4_lds[o4 * 2 + 1][0];
        float* op = out + c * HW2 + m * HW;
        for (int n = 0; n < HW; ++n) {
            float s = 0.0f;
            #pragma unroll
            for (int k3 = 0; k3 < 3; ++k3) {
                const int np = n + k3 - 1;
                if (np >= 0 && np < HW)
                    s += ta[np] * wv[k3] + tb[np] * wv[3 + k3];
            }
            op[n] = s;
        }
    }
}

extern "C" void kernel_launch(void* const* d_in, const int* in_sizes, int n_in,
                              void* d_out, int out_size, void* d_ws, size_t ws_size,
                              hipStream_t stream) {
    (void)in_sizes; (void)n_in; (void)out_size; (void)d_ws; (void)ws_size;
    const float* x  = (const float*)d_in[0];   // (1,128,56,56)
    const float* w1 = (const float*)d_in[1];   // (4,64,3)
    const float* w2 = (const float*)d_in[2];   // (32,2,3)
    float* out = (float*)d_out;                // (1,128,56,56)
    shift_conv_fused<<<HW, 128, 0, stream>>>(x, w1, w2, out);
}